// STGCNChebGraphConvProjectedGeneConnectedMultiHeadAttentionLSTMmirnaWithNumberOfConnections_71640054497509
// MI455X (gfx1250) — compile-verified
//
#include <hip/hip_runtime.h>

#define Bc   16
#define Tc   12
#define Ncn  256
#define Fc   96
#define Cce  96
#define HIDc 128
#define Hc   96
#define BnC  (Bc * Ncn)      // 4096
#define Mrows (BnC * Tc)     // 49152

typedef float v2f __attribute__((ext_vector_type(2)));
typedef float v4f __attribute__((ext_vector_type(4)));
typedef float v8f __attribute__((ext_vector_type(8)));

// ---------------------------------------------------------------------------
// WMMA fp32 GEMM (N % 32 == 0): C[M,N] = A[M,K] @ W[N,K]^T + bias, opt. ReLU.
// Each wave32 computes a 32x32 supertile (2x2 WMMA tiles): per k-step exactly
// 4 global_load_b64 + 4 V_WMMA_F32_16X16X4_F32, no branches, EXEC untouched.
// Requires M % 32 == 0 (49152 here).
// ---------------------------------------------------------------------------
__global__ void wmma_gemm_nt32(const float* __restrict__ A,
                               const float* __restrict__ W,
                               const float* __restrict__ bias,
                               float* __restrict__ C,
                               int M, int N, int K, int relu) {
  const int lane = threadIdx.x & 31;
  const int wave = threadIdx.x >> 5;
  const int ntM2 = M >> 5;
  const int ntN2 = N >> 5;
  const long st = (long)blockIdx.x * 8 + wave;
  if (st >= (long)ntM2 * ntN2) return;        // uniform per wave
  const int sm = (int)(st / ntN2);
  const int sn = (int)(st % ntN2);

  const int mrow = lane & 15;
  const int koff = (lane >> 4) * 2;           // lanes 0-15: K0,K1 ; 16-31: K2,K3
  const int row0 = sm * 32 + mrow;
  const int col0 = sn * 32 + mrow;

  const float* Ap0 = A + (long)row0 * K + koff;
  const float* Ap1 = Ap0 + (long)16 * K;
  const float* Wp0 = W + (long)col0 * K + koff;
  const float* Wp1 = Wp0 + (long)16 * K;

  const float bv0 = bias[col0];
  const float bv1 = bias[col0 + 16];
  v8f acc00, acc01, acc10, acc11;
  #pragma unroll
  for (int r = 0; r < 8; ++r) {
    acc00[r] = bv0; acc10[r] = bv0; acc01[r] = bv1; acc11[r] = bv1;
  }

  for (int k = 0; k < K; k += 4) {
    v2f a0 = *(const v2f*)(Ap0 + k);
    v2f a1 = *(const v2f*)(Ap1 + k);
    v2f b0 = *(const v2f*)(Wp0 + k);
    v2f b1 = *(const v2f*)(Wp1 + k);
    acc00 = __builtin_amdgcn_wmma_f32_16x16x4_f32(false, a0, false, b0,
                                                  (short)0, acc00, false, false);
    acc10 = __builtin_amdgcn_wmma_f32_16x16x4_f32(false, a1, false, b0,
                                                  (short)0, acc10, false, false);
    acc01 = __builtin_amdgcn_wmma_f32_16x16x4_f32(false, a0, false, b1,
                                                  (short)0, acc01, false, false);
    acc11 = __builtin_amdgcn_wmma_f32_16x16x4_f32(false, a1, false, b1,
                                                  (short)0, acc11, false, false);
  }

  const int rb  = (lane >> 4) ? 8 : 0;        // D layout: VGPR r -> row r (+8 hi lanes)
  float* Cp00 = C + (long)(sm * 32 + rb) * N + col0;
  float* Cp10 = Cp00 + (long)16 * N;
  #pragma unroll
  for (int r = 0; r < 8; ++r) {
    float v00 = acc00[r], v10 = acc10[r], v01 = acc01[r], v11 = acc11[r];
    if (relu) {
      v00 = v00 > 0.0f ? v00 : 0.0f; v10 = v10 > 0.0f ? v10 : 0.0f;
      v01 = v01 > 0.0f ? v01 : 0.0f; v11 = v11 > 0.0f ? v11 : 0.0f;
    }
    Cp00[(long)r * N]      = v00;
    Cp00[(long)r * N + 16] = v01;
    Cp10[(long)r * N]      = v10;
    Cp10[(long)r * N + 16] = v11;
  }
}

// ---------------------------------------------------------------------------
// WMMA fp32 GEMM, one 16x16 tile per wave. Used only for the tiny final
// N == 16 GEMM (z2 = z1n @ W2^T), which is a negligible share of the FLOPs.
// ---------------------------------------------------------------------------
__global__ void wmma_gemm_nt16(const float* __restrict__ A,
                               const float* __restrict__ W,
                               const float* __restrict__ bias,
                               float* __restrict__ C,
                               int M, int N, int K) {
  const int lane = threadIdx.x & 31;
  const int wave = threadIdx.x >> 5;
  const int ntN = N >> 4;
  const long tile = (long)blockIdx.x * 8 + wave;
  if (tile >= (long)(M >> 4) * ntN) return;   // uniform per wave
  const int tm = (int)(tile / ntN);
  const int tn = (int)(tile % ntN);

  const int mrow = lane & 15;
  const int koff = (lane >> 4) * 2;
  const float* Ap = A + (long)(tm * 16 + mrow) * K + koff;
  const float* Wp = W + (long)(tn * 16 + mrow) * K + koff;

  const float bv = bias[tn * 16 + mrow];
  v8f acc;
  #pragma unroll
  for (int r = 0; r < 8; ++r) acc[r] = bv;

  for (int k = 0; k < K; k += 4) {
    v2f a = *(const v2f*)(Ap + k);
    v2f b = *(const v2f*)(Wp + k);
    acc = __builtin_amdgcn_wmma_f32_16x16x4_f32(false, a, false, b,
                                                (short)0, acc, false, false);
  }

  const int r0 = tm * 16 + ((lane >> 4) ? 8 : 0);
  float* Cp = C + (long)r0 * N + tn * 16 + mrow;
  #pragma unroll
  for (int r = 0; r < 8; ++r) Cp[(long)r * N] = acc[r];
}

// ---------------------------------------------------------------------------
// LSTM recurrence: 16 sequence rows per block over all T steps.
//  pre : [Bn*T, 4H] = x@Wih^T + (bih+bhh) precomputed  (row index = bn*T + t)
//  Whh : [4H, H] (147 KB -> L2/WGP$-resident)
//  out : [Bn, T, ldOut], h written at column offset dirOff
// Per step: pre[:,t,:] staged into LDS with coalesced b128 loads, accumulators
// initialized from LDS, then 24 k-steps where each A-fragment (ds_load_b64
// from hbuf) feeds 3 WMMA column tiles.
// ---------------------------------------------------------------------------
__global__ void lstm_scan(const float* __restrict__ pre,
                          const float* __restrict__ Whh,
                          float* __restrict__ out,
                          int ldOut, int dirOff, int reverse) {
  __shared__ __align__(16) float hbuf[16][Hc];        // 6 KB
  __shared__ __align__(16) float gbuf[16][4 * Hc];    // 24 KB
  const int tid  = threadIdx.x;
  const int lane = tid & 31;
  const int wave = tid >> 5;
  const int bn0  = blockIdx.x * 16;

  __builtin_prefetch(Whh, 0, 1);        // pull Whh toward the WGP caches

  float cst[6];
  #pragma unroll
  for (int j = 0; j < 6; ++j) cst[j] = 0.0f;
  for (int p = tid; p < 16 * Hc; p += 256) hbuf[p / Hc][p % Hc] = 0.0f;
  __syncthreads();

  const int mrow  = lane & 15;
  const int koff  = (lane >> 4) * 2;
  const int rbase = (lane >> 4) * 8;
  const int colb  = wave * 48 + mrow;   // wave owns column tiles [wave*48, wave*48+48)

  for (int t = 0; t < Tc; ++t) {
    const int ti = reverse ? (Tc - 1 - t) : t;

    // ---- stage pre[:, ti, :] into gbuf (coalesced float4 loads) ----
    for (int q = tid; q < 16 * Hc; q += 256) {          // 1536 float4
      const int row = q / Hc, c4 = q % Hc;
      ((v4f*)&gbuf[row][0])[c4] =
          ((const v4f*)(pre + ((long)(bn0 + row) * Tc + ti) * (4 * Hc)))[c4];
    }
    __syncthreads();

    // ---- gate GEMM: g = pre + h @ Whh^T ----
    v8f acc[3];
    #pragma unroll
    for (int s = 0; s < 3; ++s) {
      #pragma unroll
      for (int r = 0; r < 8; ++r) acc[s][r] = gbuf[rbase + r][colb + s * 16];
    }
    for (int k = 0; k < Hc; k += 4) {
      v2f a = *(const v2f*)&hbuf[mrow][k + koff];
      #pragma unroll
      for (int s = 0; s < 3; ++s) {
        v2f b = *(const v2f*)(Whh + (long)(colb + s * 16) * Hc + koff + k);
        acc[s] = __builtin_amdgcn_wmma_f32_16x16x4_f32(false, a, false, b,
                                                       (short)0, acc[s], false, false);
      }
    }
    #pragma unroll
    for (int s = 0; s < 3; ++s) {
      #pragma unroll
      for (int r = 0; r < 8; ++r) gbuf[rbase + r][colb + s * 16] = acc[s][r];
    }
    __syncthreads();

    // ---- elementwise gates: i,f,g,o in chunks of H along the 4H axis ----
    #pragma unroll
    for (int j = 0; j < 6; ++j) {
      const int p   = tid + j * 256;
      const int row = p / Hc;
      const int hid = p % Hc;
      const float ig = gbuf[row][hid];
      const float fg = gbuf[row][Hc + hid];
      const float gg = gbuf[row][2 * Hc + hid];
      const float og = gbuf[row][3 * Hc + hid];
      const float si = 1.0f / (1.0f + __expf(-ig));
      const float sf = 1.0f / (1.0f + __expf(-fg));
      const float so = 1.0f / (1.0f + __expf(-og));
      const float cn = sf * cst[j] + si * tanhf(gg);
      cst[j] = cn;
      const float h = so * tanhf(cn);
      hbuf[row][hid] = h;
      out[((long)(bn0 + row) * Tc + ti) * ldOut + dirOff + hid] = h;
    }
    __syncthreads();
  }
}

// ---------------------------------------------------------------------------
// Elementwise / reduction kernels
// ---------------------------------------------------------------------------
__global__ void k_bias_sum(const float* __restrict__ a, const float* __restrict__ b,
                           float* __restrict__ o, int n) {
  const int i = blockIdx.x * 256 + threadIdx.x;
  if (i < n) o[i] = a[i] + b[i];
}

__global__ void k_transpose_in(const float* __restrict__ x, float* __restrict__ xl) {
  // x[B,F,T,N] -> xl[b*N+n][t][f]
  const long total = (long)Bc * Fc * Tc * Ncn;
  long i = (long)blockIdx.x * 256 + threadIdx.x;
  if (i >= total) return;
  const int n = (int)(i % Ncn);
  long r = i / Ncn;
  const int t = (int)(r % Tc); r /= Tc;
  const int f = (int)(r % Fc);
  const int b = (int)(r / Fc);
  xl[((long)(b * Ncn + n) * Tc + t) * Fc + f] = x[i];
}

__global__ void k_select(const float* __restrict__ a_std, const float* __restrict__ a_spc,
                         const int* __restrict__ conn, float* __restrict__ o, long total) {
  long i = (long)blockIdx.x * 256 + threadIdx.x;
  if (i >= total) return;
  const long row = i / (2 * Hc);
  const int n = (int)((row / Tc) % Ncn);
  o[i] = (conn[n] == 27) ? a_spc[i] : a_std[i];
}

__global__ void k_add_residual(const float* __restrict__ x, const float* __restrict__ tmp,
                               float* __restrict__ xsum, long total) {
  // xsum[b,f,t,n] = x[b,f,t,n] + tmp[((b*N+n)*T+t)*F + f]
  long i = (long)blockIdx.x * 256 + threadIdx.x;
  if (i >= total) return;
  const int n = (int)(i % Ncn);
  long r = i / Ncn;
  const int t = (int)(r % Tc); r /= Tc;
  const int f = (int)(r % Fc);
  const int b = (int)(r / Fc);
  xsum[i] = x[i] + tmp[((long)(b * Ncn + n) * Tc + t) * Fc + f];
}

__global__ void k_ln_joint(const float* __restrict__ xsum,
                           const float* __restrict__ lnw, const float* __restrict__ lnb,
                           float* __restrict__ xln) {
  // LayerNorm over joint (N,F) per (b,t); output layout [(b*T+t)*N + n][F]
  __shared__ float ssum[256];
  __shared__ float ssum2[256];
  const int b = blockIdx.x / Tc;
  const int t = blockIdx.x % Tc;
  const int tid = threadIdx.x;
  float s = 0.0f, s2 = 0.0f;
  for (int p = tid; p < Fc * Ncn; p += 256) {
    const int f = p >> 8, n = p & 255;
    const float v = xsum[(((long)(b * Fc + f)) * Tc + t) * Ncn + n];
    s += v; s2 += v * v;
  }
  ssum[tid] = s; ssum2[tid] = s2;
  __syncthreads();
  for (int st = 128; st > 0; st >>= 1) {
    if (tid < st) { ssum[tid] += ssum[tid + st]; ssum2[tid] += ssum2[tid + st]; }
    __syncthreads();
  }
  const float cnt = (float)(Fc * Ncn);
  const float m   = ssum[0] / cnt;
  const float var = ssum2[0] / cnt - m * m;
  const float inv = rsqrtf(var + 1e-5f);
  for (int p = tid; p < Fc * Ncn; p += 256) {
    const int f = p >> 8, n = p & 255;
    const float v = xsum[(((long)(b * Fc + f)) * Tc + t) * Ncn + n];
    xln[((long)(b * Tc + t) * Ncn + n) * Fc + f] =
        (v - m) * inv * lnw[n * Fc + f] + lnb[n * Fc + f];
  }
}

__global__ void k_mix(float* __restrict__ y, const float* __restrict__ a_std,
                      const float* __restrict__ a_spc, const int* __restrict__ conn,
                      long total) {
  long i = (long)blockIdx.x * 256 + threadIdx.x;
  if (i >= total) return;
  const long row = i / Cce;
  const int n = (int)(row % Ncn);
  const float a = (conn[n] == 27) ? a_spc[i] : a_std[i];
  y[i] = y[i] + 0.4f * a;
}

__global__ void k_ln_elu(const float* __restrict__ z, const float* __restrict__ w,
                         const float* __restrict__ b, float* __restrict__ o,
                         int rows, int width) {
  const int r = blockIdx.x * 256 + threadIdx.x;
  if (r >= rows) return;
  const float* zr = z + (long)r * width;
  float s = 0.0f, s2 = 0.0f;
  for (int i = 0; i < width; ++i) { const float v = zr[i]; s += v; s2 += v * v; }
  const float m   = s / width;
  const float var = s2 / width - m * m;
  const float inv = rsqrtf(var + 1e-5f);
  float* orow = o + (long)r * width;
  for (int i = 0; i < width; ++i) {
    const float v = (zr[i] - m) * inv * w[i] + b[i];
    orow[i] = v > 0.0f ? v : (__expf(v) - 1.0f);
  }
}

__global__ void k_final(const float* __restrict__ z, const float* __restrict__ W3,
                        const float* __restrict__ b3, float* __restrict__ out, int rows) {
  const int r = blockIdx.x * 256 + threadIdx.x;
  if (r >= rows) return;
  const float* zr = z + (long)r * 16;
  float acc = b3[0];
  #pragma unroll
  for (int i = 0; i < 16; ++i) acc += zr[i] * W3[i];
  out[r] = acc;   // [B,1,T,N] flat == row index (b*T+t)*N+n
}

// ---------------------------------------------------------------------------
static inline void launch_gemm(hipStream_t s, const float* A, const float* W,
                               const float* bias, float* C, int M, int N, int K,
                               int relu) {
  if ((N & 31) == 0) {
    const long st = (long)(M / 32) * (N / 32);
    const int blocks = (int)((st + 7) / 8);
    wmma_gemm_nt32<<<blocks, 256, 0, s>>>(A, W, bias, C, M, N, K, relu);
  } else {
    const long tiles = (long)(M / 16) * (N / 16);
    const int blocks = (int)((tiles + 7) / 8);
    wmma_gemm_nt16<<<blocks, 256, 0, s>>>(A, W, bias, C, M, N, K);
  }
}

extern "C" void kernel_launch(void* const* d_in, const int* in_sizes, int n_in,
                              void* d_out, int out_size, void* d_ws, size_t ws_size,
                              hipStream_t stream) {
  (void)in_sizes; (void)n_in; (void)out_size; (void)ws_size;
  const float* x    = (const float*)d_in[0];
  const int*   conn = (const int*)d_in[1];
  // input index map:
  //  2..33  : lstm_std 4 layers x {Wih_f,Whh_f,bih_f,bhh_f,Wih_b,Whh_b,bih_b,bhh_b}
  //  34..81 : lstm_spc 6 layers x same
  //  82/83 proj_W/proj_b ; 84/85 ln_w/ln_b ; 86/87 fc1 ; 88/89 fc2
  //  90..93 attn_std {in_W,in_b,out_W,out_b} ; 94..97 attn_spc ; 98..107 mlp

  float* ws = (float*)d_ws;
  size_t off = 0;
  auto A = [&](size_t n) { float* p = ws + off; off += n; return p; };
  float* xl   = A((size_t)BnC * Tc * Fc);        // [Bn,T,F]
  float* pre  = A((size_t)Mrows * 4 * Hc);       // gate preacts (reused per dir)
  float* bsum = A((size_t)4 * Hc);               // bih + bhh (reused per dir)
  float* t0   = A((size_t)BnC * Tc * 2 * Hc);
  float* t1   = A((size_t)BnC * Tc * 2 * Hc);
  float* ostd = A((size_t)BnC * Tc * 2 * Hc);
  float* ospc = A((size_t)BnC * Tc * 2 * Hc);
  float* xsum = A((size_t)Bc * Fc * Tc * Ncn);
  float* xln  = A((size_t)Mrows * Fc);
  float* h1   = A((size_t)Mrows * HIDc);         // fc1 out; later v-proj temp
  float* y    = A((size_t)Mrows * Cce);
  float* astd = A((size_t)Mrows * Cce);
  float* aspc = A((size_t)Mrows * Cce);
  float* z1   = A((size_t)Mrows * 32);
  float* z1n  = A((size_t)Mrows * 32);
  float* z2   = A((size_t)Mrows * 16);
  float* z2n  = A((size_t)Mrows * 16);

  { // x[B,F,T,N] -> xl[Bn,T,F]
    const long total = (long)Bc * Fc * Tc * Ncn;
    k_transpose_in<<<(int)((total + 255) / 256), 256, 0, stream>>>(x, xl);
  }

  auto run_dir = [&](const float* cur, int curW, const float* Wih, const float* Whh,
                     const float* bih, const float* bhh, float* dst, int dirOff,
                     int reverse) {
    k_bias_sum<<<2, 256, 0, stream>>>(bih, bhh, bsum, 4 * Hc);
    launch_gemm(stream, cur, Wih, bsum, pre, Mrows, 4 * Hc, curW, 0);
    lstm_scan<<<BnC / 16, 256, 0, stream>>>(pre, Whh, dst, 2 * Hc, dirOff, reverse);
  };

  auto run_stack = [&](int base, int layers, float* outbuf) {
    const float* cur = xl;
    int curW = Fc;
    for (int l = 0; l < layers; ++l) {
      const float* const* p = (const float* const*)(d_in + base + l * 8);
      float* dst = (l == layers - 1) ? outbuf : ((l & 1) ? t1 : t0);
      run_dir(cur, curW, p[0], p[1], p[2], p[3], dst, 0, 0);   // forward
      run_dir(cur, curW, p[4], p[5], p[6], p[7], dst, Hc, 1);  // backward
      cur = dst;
      curW = 2 * Hc;
    }
  };
  run_stack(2, 4, ostd);
  run_stack(34, 6, ospc);

  { // routed select -> t0
    const long total = (long)BnC * Tc * 2 * Hc;
    k_select<<<(int)((total + 255) / 256), 256, 0, stream>>>(ostd, ospc, conn, t0, total);
  }
  // projection: t1 = sel @ proj_W^T + proj_b   [Mrows, F], K = 2F
  launch_gemm(stream, t0, (const float*)d_in[82], (const float*)d_in[83],
              t1, Mrows, Fc, 2 * Fc, 0);
  { // residual back into [B,F,T,N]
    const long total = (long)Bc * Fc * Tc * Ncn;
    k_add_residual<<<(int)((total + 255) / 256), 256, 0, stream>>>(x, t1, xsum, total);
  }
  // joint (N,F) LayerNorm -> xln [(b*T+t)*N+n][F]
  k_ln_joint<<<Bc * Tc, 256, 0, stream>>>(xsum, (const float*)d_in[84],
                                          (const float*)d_in[85], xln);
  // fc1 + relu, fc2
  launch_gemm(stream, xln, (const float*)d_in[86], (const float*)d_in[87],
              h1, Mrows, HIDc, Fc, 1);
  launch_gemm(stream, h1, (const float*)d_in[88], (const float*)d_in[89],
              y, Mrows, Cce, HIDc, 0);
  // seq-len-1 attention == out_proj(v_proj(y)); Wv = in_W rows [2C:3C)
  launch_gemm(stream, y, (const float*)d_in[90] + 2 * Cce * Cce,
              (const float*)d_in[91] + 2 * Cce, h1, Mrows, Cce, Cce, 0);
  launch_gemm(stream, h1, (const float*)d_in[92], (const float*)d_in[93],
              astd, Mrows, Cce, Cce, 0);
  launch_gemm(stream, y, (const float*)d_in[94] + 2 * Cce * Cce,
              (const float*)d_in[95] + 2 * Cce, h1, Mrows, Cce, Cce, 0);
  launch_gemm(stream, h1, (const float*)d_in[96], (const float*)d_in[97],
              aspc, Mrows, Cce, Cce, 0);
  { // y += 0.4 * routed attention (in place)
    const long total = (long)Mrows * Cce;
    k_mix<<<(int)((total + 255) / 256), 256, 0, stream>>>(y, astd, aspc, conn, total);
  }
  // MLP head
  launch_gemm(stream, y, (const float*)d_in[98], (const float*)d_in[99],
              z1, Mrows, 32, Cce, 0);
  k_ln_elu<<<(Mrows + 255) / 256, 256, 0, stream>>>(z1, (const float*)d_in[100],
                                                    (const float*)d_in[101], z1n, Mrows, 32);
  launch_gemm(stream, z1n, (const float*)d_in[102], (const float*)d_in[103],
              z2, Mrows, 16, 32, 0);
  k_ln_elu<<<(Mrows + 255) / 256, 256, 0, stream>>>(z2, (const float*)d_in[104],
                                                    (const float*)d_in[105], z2n, Mrows, 16);
  k_final<<<(Mrows + 255) / 256, 256, 0, stream>>>(z2n, (const float*)d_in[106],
                                                   (const float*)d_in[107],
                                                   (float*)d_out, Mrows);
}